// AdaLayerNorm_28260884808105
// MI455X (gfx1250) — compile-verified
//
#include <hip/hip_runtime.h>
#include <hip/hip_bf16.h>

typedef __attribute__((ext_vector_type(2))) float v2f;
typedef __attribute__((ext_vector_type(4))) float v4f;
typedef __attribute__((ext_vector_type(8))) float v8f;

#define D_DIM  512   // normalized / output feature dim
#define WD_DIM 512   // global_feat feature dim (K of the GEMM)
#define S_SEG  64    // number of segments
#define EPSF   1e-5f

// ---------------------------------------------------------------------------
// Kernel A: g[s,d] = sum_w gf[s,w] * W[d,w] + b[d]   via V_WMMA_F32_16X16X4_F32
// One wave per 16x16 output tile. grid = (D/16, S/16) = (32, 4).
//
// A fragment (16x4 f32, 2 VGPRs/lane): lanes 0-15 -> K=0,1 ; lanes 16-31 -> K=2,3
//                                      M = lane % 16
// B fragment (4x16 f32, 2 VGPRs/lane): mirrored K striping, N = lane % 16.
// B[w, dbase+n] = W[dbase+n, w]  -> column n of B is row (dbase+n) of W,
// contiguous in memory, so each lane's 2 K-values are one float2 load.
// C/D (16x16 f32, 8 VGPRs): VGPR v -> M = v + (lane/16)*8, N = lane % 16.
// ---------------------------------------------------------------------------
__global__ __launch_bounds__(32) void ada_g_gemm(const float* __restrict__ gf,
                                                 const float* __restrict__ Wm,
                                                 const float* __restrict__ bias,
                                                 float* __restrict__ g) {
    const int lane  = threadIdx.x;          // 0..31 (wave32)
    const int dbase = blockIdx.x * 16;      // output column tile
    const int sbase = blockIdx.y * 16;      // output row tile
    const int mn    = lane & 15;            // M for A, N for B/C
    const int koff  = (lane >> 4) * 2;      // K sub-offset within the 4-wide tile

    const float* aptr = gf + (size_t)(sbase + mn) * WD_DIM + koff;
    const float* bptr = Wm + (size_t)(dbase + mn) * WD_DIM + koff;

    v8f c = {};
    #pragma unroll 4
    for (int k = 0; k < WD_DIM; k += 4) {
        v2f a  = *(const v2f*)(aptr + k);
        v2f bb = *(const v2f*)(bptr + k);
        // 8 args: (neg_a, A, neg_b, B, c_mod, C, reuse_a, reuse_b)
        c = __builtin_amdgcn_wmma_f32_16x16x4_f32(false, a, false, bb,
                                                  (short)0, c, false, false);
    }

    const float bv   = bias[dbase + mn];    // b[d] broadcast over rows
    const int   mrow = (lane >> 4) * 8;
    #pragma unroll
    for (int v = 0; v < 8; ++v) {
        g[(size_t)(sbase + mrow + v) * D_DIM + dbase + mn] = c[v] + bv;
    }
}

// ---------------------------------------------------------------------------
// Kernel B: per-row LayerNorm * gathered g row.  One wave32 per row of feat.
// 256 threads/block = 8 waves = 8 rows per block.  16384 blocks.
// Each lane owns 16 floats of the row as 4x float4 (fully coalesced b128).
// feat / out use non-temporal hints (streamed once, 256 MB each) so the
// 128 KB g table stays hot in WGP$/L2.
// ---------------------------------------------------------------------------
__global__ __launch_bounds__(256) void ada_ln_apply(const float* __restrict__ feat,
                                                    const long long* __restrict__ offset,
                                                    const float* __restrict__ g,
                                                    float* __restrict__ out) {
    const int  lane = threadIdx.x & 31;
    const int  wave = threadIdx.x >> 5;
    const long long row = (long long)blockIdx.x * 8 + wave;

    // searchsorted(offset, row, 'right'): first s with offset[s] > row.
    // 6-step binary search, done redundantly per lane (64-entry table in L2).
    int lo = 0, hi = S_SEG - 1;
    while (lo < hi) {
        int mid = (lo + hi) >> 1;
        if (offset[mid] > row) hi = mid; else lo = mid + 1;
    }
    const int seg = lo;

    const float* frow = feat + (size_t)row * D_DIM;
    v4f x[4];
    float sum = 0.f, sq = 0.f;
    #pragma unroll
    for (int c = 0; c < 4; ++c) {
        x[c] = __builtin_nontemporal_load((const v4f*)(frow + c * 128 + lane * 4));
        #pragma unroll
        for (int j = 0; j < 4; ++j) {
            sum += x[c][j];
            sq   = fmaf(x[c][j], x[c][j], sq);
        }
    }
    // wave32 butterfly reduction
    #pragma unroll
    for (int off = 16; off > 0; off >>= 1) {
        sum += __shfl_xor(sum, off, 32);
        sq  += __shfl_xor(sq,  off, 32);
    }

    const float inv  = 1.0f / (float)D_DIM;
    const float mean = sum * inv;
    const float var  = fmaf(sq, inv, -mean * mean);   // E[x^2] - mean^2
    const float rs   = rsqrtf(var + EPSF);

    const float* grow = g   + (size_t)seg * D_DIM;    // hot in cache
    float*       orow = out + (size_t)row * D_DIM;
    #pragma unroll
    for (int c = 0; c < 4; ++c) {
        v4f gv = *(const v4f*)(grow + c * 128 + lane * 4);
        v4f o;
        #pragma unroll
        for (int j = 0; j < 4; ++j) {
            o[j] = gv[j] * ((x[c][j] - mean) * rs);
        }
        __builtin_nontemporal_store(o, (v4f*)(orow + c * 128 + lane * 4));
    }
}

extern "C" void kernel_launch(void* const* d_in, const int* in_sizes, int n_in,
                              void* d_out, int out_size, void* d_ws, size_t ws_size,
                              hipStream_t stream) {
    const float*     feat   = (const float*)d_in[0];      // [N, 512] f32
    const float*     gfeat  = (const float*)d_in[1];      // [64, 512] f32
    const long long* offset = (const long long*)d_in[2];  // [64] int64
    const float*     Wm     = (const float*)d_in[3];      // [512, 512] f32
    const float*     bias   = (const float*)d_in[4];      // [512] f32
    float*           out    = (float*)d_out;              // [N, 512] f32
    float*           g      = (float*)d_ws;               // 64*512 f32 = 128 KB scratch

    // Kernel A: tiny WMMA GEMM producing g = gf @ W^T + b
    dim3 gridA(D_DIM / 16, S_SEG / 16);                   // 32 x 4 tiles
    ada_g_gemm<<<gridA, 32, 0, stream>>>(gfeat, Wm, bias, g);

    // Kernel B: bandwidth-bound LayerNorm * gather-broadcast
    const int N = in_sizes[0] / D_DIM;                    // 131072
    ada_ln_apply<<<N / 8, 256, 0, stream>>>(feat, offset, g, out);
}